// PCNLayer_62612033241754
// MI455X (gfx1250) — compile-verified
//
#include <hip/hip_runtime.h>

typedef __attribute__((ext_vector_type(2))) float v2f;
typedef __attribute__((ext_vector_type(8))) float v8f;

namespace {
constexpr int NB   = 32;                     // batch N
constexpr int HH   = 16;                     // H
constexpr int CIN  = 32;                     // Cin
constexpr int COUT = 64;                     // Cout
constexpr int S    = 64;
constexpr int P    = 4;
constexpr int LT   = 12;                     // frames per trajectory
constexpr int KW   = 2;                      // kernel width
constexpr int STEP = 2;                      // dilation
constexpr int LO   = LT - STEP * (KW - 1);   // 10
constexpr int Q    = HH * LO;                // 160 positions per (s, n)
constexpr int QT   = Q / 16;                 // 10 position tiles
constexpr int COT  = COUT / 16;              // 4 Cout tiles
constexpr int WAVES = S * NB * QT * COT;     // 81920 waves
constexpr int WPB  = 4;                      // waves per block (one full Cout per block)
}

// One wave32 computes a 16(Cout) x 16(position) output tile of
// out[n, s, co, h, lo] = mean_p( conv1 * sigmoid(conv2) ), using
// V_WMMA_F32_16X16X4_F32 with K = Cin*KW = 64 reduced in 16 steps.
__global__ __launch_bounds__(WPB * 32) void pcn_glu_conv_wmma(
    const float* __restrict__ x,   // [T, N, H, Cin]
    const float* __restrict__ w1,  // [Cout, Cin, 1, KW] -> row of 64 contiguous floats
    const float* __restrict__ b1,  // [Cout]
    const float* __restrict__ w2,
    const float* __restrict__ b2,
    const int*   __restrict__ traj, // [S, P, LT]
    float* __restrict__ out)        // [N, S, Cout, H, LO]
{
  const int lane = threadIdx.x & 31;
  const int wid  = blockIdx.x * WPB + (threadIdx.x >> 5);

  const int cot = wid % COT;
  int tmp       = wid / COT;
  const int qt  = tmp % QT;  tmp /= QT;
  const int n   = tmp % NB;
  const int s   = tmp / NB;

  const int hi  = lane >> 4;    // lane half: selects K pair (A/B) and C row offset
  const int col = lane & 15;    // B/C column == A row

  const int q   = qt * 16 + col;  // flattened (h, lo) position
  const int h   = q / LO;
  const int lo  = q % LO;
  const int co_base = cot * 16;

  // ---- Preload weight (A) fragments for both convs -------------------------
  // A[m = col][k = kb*4 + hi*2 + {0,1}];  w row index co*64 + (ci*2 + kk) = co*64 + k
  const float* w1p = w1 + (co_base + col) * (CIN * KW) + hi * 2;
  const float* w2p = w2 + (co_base + col) * (CIN * KW) + hi * 2;
  v2f a1[16], a2[16];
#pragma unroll
  for (int kb = 0; kb < 16; ++kb) {
    a1[kb] = *(const v2f*)(w1p + kb * 4);   // 64-bit loads
    a2[kb] = *(const v2f*)(w2p + kb * 4);
  }

  // Bias values for the 8 C rows this lane owns (row M = v + hi*8)
  float bb1[8], bb2[8];
#pragma unroll
  for (int v = 0; v < 8; ++v) {
    const int co = co_base + v + hi * 8;
    bb1[v] = b1[co];
    bb2[v] = b2[co];
  }

  v8f acc = {};   // running sum over trajectories (mean/P applied at store)
  for (int p = 0; p < P; ++p) {
    const int t0 = traj[(s * P + p) * LT + lo];         // kernel tap 0: l = lo
    const int t1 = traj[(s * P + p) * LT + lo + STEP];  // kernel tap 1: l = lo + 2
    const float* xb0 = x + (((size_t)t0 * NB + n) * HH + h) * CIN;
    const float* xb1 = x + (((size_t)t1 * NB + n) * HH + h) * CIN;

    v8f c1 = {}, c2 = {};
#pragma unroll
    for (int kb = 0; kb < 16; ++kb) {
      const int ci = kb * 2 + hi;    // B row K = kb*4 + hi*2 + kk  ->  ci = K>>1, kk = K&1
      v2f b;
      b.x = xb0[ci];
      b.y = xb1[ci];
      // 8 args: (neg_a, A, neg_b, B, c_mod, C, reuse_a, reuse_b)
      c1 = __builtin_amdgcn_wmma_f32_16x16x4_f32(false, a1[kb], false, b,
                                                 (short)0, c1, false, false);
      c2 = __builtin_amdgcn_wmma_f32_16x16x4_f32(false, a2[kb], false, b,
                                                 (short)0, c2, false, false);
    }

    // gate: conv1 * sigmoid(conv2), then accumulate for the trajectory mean
#pragma unroll
    for (int v = 0; v < 8; ++v) {
      const float y1 = c1[v] + bb1[v];
      const float y2 = c2[v] + bb2[v];
      acc[v] += y1 * (1.0f / (1.0f + __expf(-y2)));
    }
  }

  // store: out[n, s, co_base + hi*8 + v, h, lo];  co stride = H*LO floats
  float* op = out + ((((size_t)n * S + s) * COUT + co_base + hi * 8) * HH + h) * LO + lo;
#pragma unroll
  for (int v = 0; v < 8; ++v) {
    op[(size_t)v * (HH * LO)] = acc[v] * 0.25f;
  }
}

extern "C" void kernel_launch(void* const* d_in, const int* in_sizes, int n_in,
                              void* d_out, int out_size, void* d_ws, size_t ws_size,
                              hipStream_t stream) {
  const float* x    = (const float*)d_in[0];
  const float* w1   = (const float*)d_in[1];
  const float* b1   = (const float*)d_in[2];
  const float* w2   = (const float*)d_in[3];
  const float* b2   = (const float*)d_in[4];
  const int*   traj = (const int*)d_in[5];
  float* out = (float*)d_out;

  dim3 grid(WAVES / WPB);    // 20480 blocks
  dim3 block(WPB * 32);      // 128 threads = 4 waves
  hipLaunchKernelGGL(pcn_glu_conv_wmma, grid, block, 0, stream,
                     x, w1, b1, w2, b2, traj, out);
}